// E8P12_codebook_45990509806220
// MI455X (gfx1250) — compile-verified
//
#include <hip/hip_runtime.h>

typedef float v2f __attribute__((ext_vector_type(2)));
typedef float v8f __attribute__((ext_vector_type(8)));

#define WAVES_PER_BLOCK 8
#define PTS_PER_WAVE    16
#define PTS_PER_BLOCK   (WAVES_PER_BLOCK * PTS_PER_WAVE)

// E8P12 codebook quantization.
// Score GEMM: S = (2*Ya)(16x8) @ gridT(8x16chunk) + Cinit(-||g||^2), via two
// chained V_WMMA_F32_16X16X4_F32 per pass per chunk. B fragments and the
// -norm C-initializer are shared by the (X+0.25) and (X-0.25) passes.
__global__ __launch_bounds__(256) void e8p12_wmma_kernel(
    const float* __restrict__ X,        // (N, 8)
    const float* __restrict__ grid,     // (G, 8)
    const float* __restrict__ gnorm,    // (G,)
    const int*   __restrict__ allcombo, // (128, G)
    const int*   __restrict__ idx_map,  // (256,)
    float* __restrict__ out_vals,       // (N, 8)
    float* __restrict__ out_idx,        // (N,)  int16 value stored as float
    int N, int G)
{
    __shared__ float sS[2][WAVES_PER_BLOCK][16];
    __shared__ int   sI[2][WAVES_PER_BLOCK][16];

    const int tid  = threadIdx.x;
    const int wave = tid >> 5;
    const int lane = tid & 31;
    const int m    = lane & 15;          // tile row M (and tile col N for C/D)
    const int kb   = (lane >> 4) * 2;    // K base of this lane's A/B fragment

    const int pbase = blockIdx.x * PTS_PER_BLOCK + wave * PTS_PER_WAVE;
    const int p     = pbase + m;
    const int p_ld  = (p < N) ? p : (N - 1);

    // ---- Load X row (each half-wave duplicates; L0/L2 resident) ----
    float x[8];
    {
        const float* xr = X + (size_t)p_ld * 8;
        #pragma unroll
        for (int j = 0; j < 8; ++j) x[j] = xr[j];
    }

    // ---- Build Ya (sign-canonicalized abs) for both passes ----
    float yaP[8], yaM[8];
    float n2P = 0.f, n2M = 0.f;     // ||Y||^2 per pass
    int mbP = 0, mbM = 0;           // bit j set <=> mask_j == -1
    int mintP = 0, mintM = 0;       // sum_j (mask_j<0) * (128>>j)
    {
        int cp = 0, cm = 0;
        #pragma unroll
        for (int j = 0; j < 8; ++j) {
            float yp = x[j] + 0.25f;
            float ym = x[j] - 0.25f;
            n2P = fmaf(yp, yp, n2P);
            n2M = fmaf(ym, ym, n2M);
            int np_ = (yp < 0.f) ? 1 : 0;
            int nm_ = (ym < 0.f) ? 1 : 0;
            cp += np_; cm += nm_;
            mbP |= np_ << j; mbM |= nm_ << j;
            yaP[j] = fabsf(yp); yaM[j] = fabsf(ym);
        }
        int oddP = cp & 1, oddM = cm & 1;
        if (oddP) yaP[0] = -yaP[0];
        if (oddM) yaM[0] = -yaM[0];
        mbP ^= oddP;   // flips only bit 0 (mask_0 sign toggled by odd parity)
        mbM ^= oddM;
        #pragma unroll
        for (int j = 0; j < 8; ++j) {
            mintP |= ((mbP >> j) & 1) << (7 - j);
            mintM |= ((mbM >> j) & 1) << (7 - j);
        }
    }

    // ---- A fragments, pre-scaled by 2 (exact: power-of-two scaling).
    //      ISA 16x4 f32 A layout: lane L row M=L%16, VGPR0/1 = K {kb, kb+1};
    //      second WMMA covers K {kb+4, kb+5}. ----
    v2f aPl, aPh, aMl, aMh;
    aPl[0] = 2.f * yaP[kb];     aPl[1] = 2.f * yaP[kb + 1];
    aPh[0] = 2.f * yaP[kb + 4]; aPh[1] = 2.f * yaP[kb + 5];
    aMl[0] = 2.f * yaM[kb];     aMl[1] = 2.f * yaM[kb + 1];
    aMh[0] = 2.f * yaM[kb + 4]; aMh[1] = 2.f * yaM[kb + 5];

    // ---- Running argmax over all G columns ----
    float bSP[8], bSM[8];
    int   bIP[8], bIM[8];
    #pragma unroll
    for (int v = 0; v < 8; ++v) {
        bSP[v] = -__builtin_inff(); bSM[v] = -__builtin_inff();
        bIP[v] = 0; bIM[v] = 0;
    }

    const int nFull   = G >> 4;      // guard-free chunks
    const int hasTail = G & 15;

    // Uniform bases advance in SALU; lane-dependent offsets are invariant, so
    // loads stay in saddr+voffset form with scalar increments.
    const int voffB = m * 8 + kb;    // element offset within the chunk tile
    const float* gb  = grid;
    const float* gnb = gnorm;
    int n = m;                       // this lane's current column / grid row

    // ---- Main loop: no bounds checks, no clamps ----
    #pragma unroll 2
    for (int c = 0; c < nFull; ++c) {
        // B fragment: lane L column N=L%16, same K striping as A.
        v2f bl, bh;
        bl[0] = gb[voffB];     bl[1] = gb[voffB + 1];
        bh[0] = gb[voffB + 4]; bh[1] = gb[voffB + 5];

        // C initializer = -||g_n||^2, identical across the lane's 8 rows and
        // shared by both passes.
        const float negNrm = -gnb[m];
        v8f cInit;
        #pragma unroll
        for (int v = 0; v < 8; ++v) cInit[v] = negNrm;

        v8f accP, accM;
        accP = __builtin_amdgcn_wmma_f32_16x16x4_f32(false, aPl, false, bl,
                                                     (short)0, cInit, false, false);
        accP = __builtin_amdgcn_wmma_f32_16x16x4_f32(false, aPh, false, bh,
                                                     (short)0, accP, false, false);
        accM = __builtin_amdgcn_wmma_f32_16x16x4_f32(false, aMl, false, bl,
                                                     (short)0, cInit, false, false);
        accM = __builtin_amdgcn_wmma_f32_16x16x4_f32(false, aMh, false, bh,
                                                     (short)0, accM, false, false);

        #pragma unroll
        for (int v = 0; v < 8; ++v) {
            float sP = accP[v];
            float sM = accM[v];
            if (sP > bSP[v]) { bSP[v] = sP; bIP[v] = n; }
            if (sM > bSM[v]) { bSM[v] = sM; bIM[v] = n; }
        }

        gb  += 128;   // 16 rows * 8 floats
        gnb += 16;
        n   += 16;
    }

    // ---- Tail chunk (uniform branch; EXEC stays full) ----
    if (hasTail) {
        const int n_ld = (n < G) ? n : (G - 1);

        const float* gr = grid + (size_t)n_ld * 8;
        v2f bl, bh;
        bl[0] = gr[kb];     bl[1] = gr[kb + 1];
        bh[0] = gr[kb + 4]; bh[1] = gr[kb + 5];

        float nl = gnorm[n_ld];
        const float negNrm = (n < G) ? -nl : -__builtin_inff();
        v8f cInit;
        #pragma unroll
        for (int v = 0; v < 8; ++v) cInit[v] = negNrm;

        v8f accP, accM;
        accP = __builtin_amdgcn_wmma_f32_16x16x4_f32(false, aPl, false, bl,
                                                     (short)0, cInit, false, false);
        accP = __builtin_amdgcn_wmma_f32_16x16x4_f32(false, aPh, false, bh,
                                                     (short)0, accP, false, false);
        accM = __builtin_amdgcn_wmma_f32_16x16x4_f32(false, aMl, false, bl,
                                                     (short)0, cInit, false, false);
        accM = __builtin_amdgcn_wmma_f32_16x16x4_f32(false, aMh, false, bh,
                                                     (short)0, accM, false, false);

        #pragma unroll
        for (int v = 0; v < 8; ++v) {
            float sP = accP[v];
            float sM = accM[v];
            if (sP > bSP[v]) { bSP[v] = sP; bIP[v] = n; }
            if (sM > bSM[v]) { bSM[v] = sM; bIM[v] = n; }
        }
    }

    // ---- Butterfly argmax within each 16-lane half (C-layout halves).
    //      Tie-break: smaller index (matches jnp.argmax first-occurrence). ----
    #pragma unroll
    for (int msk = 1; msk <= 8; msk <<= 1) {
        #pragma unroll
        for (int v = 0; v < 8; ++v) {
            float oS = __shfl_xor(bSP[v], msk, 32);
            int   oI = __shfl_xor(bIP[v], msk, 32);
            if (oS > bSP[v] || (oS == bSP[v] && oI < bIP[v])) { bSP[v] = oS; bIP[v] = oI; }
            oS = __shfl_xor(bSM[v], msk, 32);
            oI = __shfl_xor(bIM[v], msk, 32);
            if (oS > bSM[v] || (oS == bSM[v] && oI < bIM[v])) { bSM[v] = oS; bIM[v] = oI; }
        }
    }

    // Lane 0 holds results for points M=0..7 (vgpr v), lane 16 for M=8..15.
    if (lane == 0 || lane == 16) {
        int half = lane >> 4;
        #pragma unroll
        for (int v = 0; v < 8; ++v) {
            sS[0][wave][half * 8 + v] = bSP[v];
            sI[0][wave][half * 8 + v] = bIP[v];
            sS[1][wave][half * 8 + v] = bSM[v];
            sI[1][wave][half * 8 + v] = bIM[v];
        }
    }
    __syncthreads();

    // ---- Epilogue: one lane per point ----
    if (lane < 16 && p < N) {
        float scP = sS[0][wave][m]; int qP = sI[0][wave][m];
        float scM = sS[1][wave][m]; int qM = sI[1][wave][m];

        // err^2 = ||Y||^2 - (2 Ya.g - ||g||^2)
        float e2P = n2P - scP;
        float e2M = n2M - scM;
        bool which = e2P < e2M;

        int   q    = which ? qP    : qM;
        int   mint = which ? mintP : mintM;
        int   mb   = which ? mbP   : mbM;
        float off  = which ? -0.25f : 0.25f;

        const float* gq = grid + (size_t)q * 8;
        float* ov = out_vals + (size_t)p * 8;
        #pragma unroll
        for (int j = 0; j < 8; ++j) {
            float s = ((mb >> j) & 1) ? -1.f : 1.f;
            ov[j] = fmaf(gq[j], s, off);
        }

        int row  = idx_map[mint];
        int ridx = allcombo[(size_t)row * G + q];
        // grid_idx_map[k] == int16(k - 32768); k = which ? ridx+32768 : ridx
        int k16 = which ? ridx : (ridx ^ 0x8000);
        short idx16 = (short)(k16 & 0xFFFF);
        out_idx[p] = (float)idx16;
    }
}

extern "C" void kernel_launch(void* const* d_in, const int* in_sizes, int n_in,
                              void* d_out, int out_size, void* d_ws, size_t ws_size,
                              hipStream_t stream) {
    const float* X        = (const float*)d_in[0];
    const float* grid     = (const float*)d_in[1];
    const float* gnorm    = (const float*)d_in[2];
    const int*   allcombo = (const int*)d_in[3];
    const int*   idx_map  = (const int*)d_in[4];
    // d_in[5] (int_map) and d_in[6] (grid_idx_map) are folded in analytically.

    const int N = in_sizes[0] / 8;
    const int G = in_sizes[1] / 8;

    float* out_vals = (float*)d_out;
    float* out_idx  = (float*)d_out + (size_t)N * 8;

    const int blocks = (N + PTS_PER_BLOCK - 1) / PTS_PER_BLOCK;
    e8p12_wmma_kernel<<<blocks, 256, 0, stream>>>(
        X, grid, gnorm, allcombo, idx_map, out_vals, out_idx, N, G);
}